// GriddingDistance_128849019469
// MI455X (gfx1250) — compile-verified
//
#include <hip/hip_runtime.h>

#define WG 256
#define D 128
#define D3 (128 * 128 * 128)

// ---------------------------------------------------------------------------
// Zero the 64MB output using the CDNA5 async LDS->global store engine.
// A 4KB block of LDS is zeroed once, then each lane streams b128 async stores
// (ASYNCcnt-tracked, no VGPR data movement). HW stalls issue on counter
// saturation, so no manual throttling is needed; S_ENDPGM implies wait-idle,
// and we also wait explicitly.
// ---------------------------------------------------------------------------
__global__ void __launch_bounds__(WG) zero_async_kernel(float4* __restrict__ out,
                                                        long n4) {
    __shared__ float4 zbuf[WG];
    zbuf[threadIdx.x] = make_float4(0.f, 0.f, 0.f, 0.f);
    __syncthreads();

    // Generic pointer to __shared__ is {SHARED_BASE, lds_offset}; low 32 bits
    // are the workgroup-relative LDS byte address the async op wants in VDATA.
    unsigned lds_off = (unsigned)(unsigned long long)&zbuf[threadIdx.x];

    long stride = (long)gridDim.x * WG;
    for (long i = (long)blockIdx.x * WG + threadIdx.x; i < n4; i += stride) {
        const float4* dst = out + i;
        asm volatile("global_store_async_from_lds_b128 %0, %1, off"
                     :: "v"(dst), "v"(lds_off)
                     : "memory");
    }
    asm volatile("s_wait_asynccnt 0" ::: "memory");
}

// ---------------------------------------------------------------------------
// Trilinear splat: one thread per point, 8 native non-returning f32 atomics.
// Point loads use NT temporal hint (streamed once; keep L2 for the grids).
// ---------------------------------------------------------------------------
__global__ void __launch_bounds__(WG) splat_kernel(const float* __restrict__ pred,
                                                   const float* __restrict__ gt,
                                                   float* __restrict__ out,
                                                   int N, int B) {
    int pt = blockIdx.x * WG + threadIdx.x;
    if (pt >= N) return;
    const int b     = blockIdx.y;
    const int cloud = blockIdx.z;

    const float* __restrict__ src = cloud ? gt : pred;
    const float* p = src + ((long)b * N + pt) * 3;

    // scale = D/2 = 64 for every dim; mins = -64
    float px = __builtin_nontemporal_load(p + 0) * 64.0f;
    float py = __builtin_nontemporal_load(p + 1) * 64.0f;
    float pz = __builtin_nontemporal_load(p + 2) * 64.0f;

    float lx = floorf(px), ly = floorf(py), lz = floorf(pz);
    float fx = px - lx, fy = py - ly, fz = pz - lz;
    int ix = (int)lx + 64, iy = (int)ly + 64, iz = (int)lz + 64;

    float wx[2] = {1.0f - fx, fx};
    float wy[2] = {1.0f - fy, fy};
    float wz[2] = {1.0f - fz, fz};

    float* __restrict__ base = out + ((long)cloud * B + b) * (long)D3;

#pragma unroll
    for (int cx = 0; cx < 2; ++cx) {
        unsigned x = (unsigned)(ix + cx);
        if (x >= (unsigned)D) continue;          // per-corner validity (ref parity)
#pragma unroll
        for (int cy = 0; cy < 2; ++cy) {
            unsigned y = (unsigned)(iy + cy);
            if (y >= (unsigned)D) continue;
            float    wxy     = wx[cx] * wy[cy];
            unsigned rowbase = (x * D + y) * D;
#pragma unroll
            for (int cz = 0; cz < 2; ++cz) {
                unsigned z = (unsigned)(iz + cz);
                if (z >= (unsigned)D) continue;
                // native global_atomic_add_f32, non-returning (STOREcnt path)
                unsafeAtomicAdd(base + (rowbase + z), wxy * wz[cz]);
            }
        }
    }
}

extern "C" void kernel_launch(void* const* d_in, const int* in_sizes, int n_in,
                              void* d_out, int out_size, void* d_ws, size_t ws_size,
                              hipStream_t stream) {
    (void)d_ws; (void)ws_size;
    const float* pred = (const float*)d_in[0];
    const float* gt   = (const float*)(n_in > 1 ? d_in[1] : d_in[0]);
    float*       out  = (float*)d_out;

    int B = out_size / (2 * D3);           // pred_grid + gt_grid concatenated
    if (B < 1) B = 1;
    int N = in_sizes[0] / (3 * B);

    // Pass 1: zero 64MB of output via async LDS->global b128 stores.
    long n4    = (long)out_size / 4;       // float4 count (out_size % 4 == 0)
    long nblk  = (n4 + WG - 1) / WG;
    int  zgrid = (int)(nblk < 4096 ? nblk : 4096);
    if (zgrid > 0) {
        zero_async_kernel<<<zgrid, WG, 0, stream>>>((float4*)out, n4);
    }

    // Pass 2: splat both clouds in one launch: grid = (points, batch, cloud).
    dim3 grid((unsigned)((N + WG - 1) / WG), (unsigned)B, 2u);
    splat_kernel<<<grid, WG, 0, stream>>>(pred, gt, out, N, B);
}